// ContrastiveLoss_65240553226776
// MI455X (gfx1250) — compile-verified
//
#include <hip/hip_runtime.h>
#include <math.h>

// Problem constants (from reference setup_inputs)
#define N_PTS 8192
#define DIM   128
#define TINV  5.0f      // 1/T, T=0.2
#define EPS_C 1e-5f

typedef __attribute__((ext_vector_type(2))) float v2f;
typedef __attribute__((ext_vector_type(8))) float v8f;

// d_ws layout in floats:
//   [0, N)        : norms
//   [N, 2N)       : neg row sums
//   [2N, 2N+512)  : per-block (loss, count) partials for pass 2
#define WS_NORM 0
#define WS_NEG  (N_PTS)
#define WS_PART (2*N_PTS)
#define NBLK    (N_PTS/32)   // 256 blocks for the pass kernels

// ---- gfx1250 async global->LDS copy (ASYNCcnt path, bypasses VGPRs) --------
__device__ __forceinline__ void async_copy16(const void* g, void* l) {
  unsigned ldsOff = (unsigned)(size_t)l;   // low 32 bits of flat LDS addr = LDS offset
  asm volatile("global_load_async_to_lds_b128 %0, %1, off"
               :: "v"(ldsOff), "v"(g) : "memory");
}
__device__ __forceinline__ void async_wait0() {
  asm volatile("s_wait_asynccnt 0x0" ::: "memory");
}

// ---------------- Kernel 1: row norms (wave per row, coalesced) -------------
__global__ __launch_bounds__(256)
void norms_kernel(const float* __restrict__ x, float* __restrict__ ws) {
  int row  = (blockIdx.x * blockDim.x + threadIdx.x) >> 5;
  int lane = threadIdx.x & 31;
  const float* r = x + (size_t)row * DIM;
  float s = 0.f;
  #pragma unroll
  for (int i = 0; i < DIM; i += 32) { float v = r[i + lane]; s += v * v; }
  #pragma unroll
  for (int m = 16; m >= 1; m >>= 1) s += __shfl_xor(s, m, 32);
  if (lane == 0) ws[WS_NORM + row] = sqrtf(s);
}

// ------------- Kernels 2/3: fused WMMA Gram-matrix passes -------------------
// PASS 0: neg_i = sum over different-label j of exp(sim_ij/T)
// PASS 1: loss terms over same-label off-diagonal pairs + pair count
template <int PASS>
__global__ __launch_bounds__(256)
void pass_kernel(const float* __restrict__ x, const int* __restrict__ lab,
                 float* __restrict__ ws) {
  __shared__ float Alds[32 * 132];     // 32 rows, padded to 132 floats (bank-clean)
  __shared__ float Blds[64 * 132];     // 64 cols, padded
  __shared__ float negPart[4][32];
  __shared__ float wavePart[8][2];

  const int tid    = threadIdx.x;
  const int lane   = tid & 31;
  const int wave   = tid >> 5;
  const int rowSub = wave & 1;         // which 16-row subtile (0..1)
  const int colSub = wave >> 1;        // which 16-col subtile (0..3)
  const int rBase  = blockIdx.x * 32;

  const int hi   = lane >> 4;          // 0: lanes 0-15, 1: lanes 16-31
  const int nIdx = lane & 15;

  // Stage A block (32x128) into LDS via async-to-LDS B128 copies
  for (int idx = tid; idx < 32 * 32; idx += 256) {
    int r = idx >> 5, q = idx & 31;
    async_copy16(&x[(size_t)(rBase + r) * DIM + q * 4], &Alds[r * 132 + q * 4]);
  }

  // Row-side constants held in registers: lane owns rows m = v + 8*hi
  float normR[8], negR[8];
  int   labR[8], rowG[8];
  #pragma unroll
  for (int v = 0; v < 8; ++v) {
    int i = rBase + rowSub * 16 + v + 8 * hi;
    rowG[v]  = i;
    normR[v] = ws[WS_NORM + i];
    labR[v]  = lab[i];
    negR[v]  = (PASS == 1) ? ws[WS_NEG + i] : 0.f;
  }

  float accNeg[8];
  #pragma unroll
  for (int v = 0; v < 8; ++v) accNeg[v] = 0.f;
  float lossAcc = 0.f, cntAcc = 0.f;

  const int rA  = rowSub * 16 + nIdx;   // A fragment row in LDS
  const int rB  = colSub * 16 + nIdx;   // B fragment row in LDS
  const int off = hi * 2;               // K offset per ISA 16x4 f32 layout

  #define LDA(kk) (*reinterpret_cast<const v2f*>(&Alds[rA * 132 + (kk) + off]))
  #define LDB(kk) (*reinterpret_cast<const v2f*>(&Blds[rB * 132 + (kk) + off]))

  for (int cBase = 0; cBase < N_PTS; cBase += 64) {
    __syncthreads();     // all waves finished reading previous B chunk
    // Stage B block (64x128) into LDS via async copies
    for (int idx = tid; idx < 64 * 32; idx += 256) {
      int r = idx >> 5, q = idx & 31;
      async_copy16(&x[(size_t)(cBase + r) * DIM + q * 4], &Blds[r * 132 + q * 4]);
    }
    async_wait0();       // this wave's async copies (incl. A on first iter) done
    __syncthreads();     // all waves' copies visible

    // 16x16 tile, K=128: 32 x V_WMMA_F32_16X16X4_F32 in 4 independent chains
    v8f c0 = {}, c1 = {}, c2 = {}, c3 = {};
    #pragma unroll
    for (int k = 0; k < DIM; k += 16) {
      c0 = __builtin_amdgcn_wmma_f32_16x16x4_f32(false, LDA(k),      false, LDB(k),      (short)0, c0, false, false);
      c1 = __builtin_amdgcn_wmma_f32_16x16x4_f32(false, LDA(k + 4),  false, LDB(k + 4),  (short)0, c1, false, false);
      c2 = __builtin_amdgcn_wmma_f32_16x16x4_f32(false, LDA(k + 8),  false, LDB(k + 8),  (short)0, c2, false, false);
      c3 = __builtin_amdgcn_wmma_f32_16x16x4_f32(false, LDA(k + 12), false, LDB(k + 12), (short)0, c3, false, false);
    }
    v8f c = (c0 + c1) + (c2 + c3);

    // Epilogue: lane holds C[m][n], m = v + 8*hi, n = nIdx
    int   jG    = cBase + colSub * 16 + nIdx;
    int   labC  = lab[jG];
    float normC = ws[WS_NORM + jG];
    #pragma unroll
    for (int v = 0; v < 8; ++v) {
      float dot   = c[v];
      float denom = fmaxf(normR[v] * normC, EPS_C);
      float su    = (dot / denom) * TINV;      // sim / T
      if (PASS == 0) {
        if (labR[v] != labC) accNeg[v] += __expf(su);
      } else {
        if (labR[v] == labC && rowG[v] != jG) {
          float e = __expf(su);
          lossAcc += __logf(e + negR[v]) - su;  // -log(e/(e+neg))
          cntAcc  += 1.f;
        }
      }
    }
  }
  #undef LDA
  #undef LDB

  if (PASS == 0) {
    // Reduce over the 16 lanes (N dimension) inside each half-wave
    #pragma unroll
    for (int v = 0; v < 8; ++v) {
      float s = accNeg[v];
      #pragma unroll
      for (int m = 8; m >= 1; m >>= 1) s += __shfl_xor(s, m, 16);
      accNeg[v] = s;
    }
    if (nIdx == 0) {
      #pragma unroll
      for (int v = 0; v < 8; ++v)
        negPart[colSub][rowSub * 16 + v + 8 * hi] = accNeg[v];
    }
    __syncthreads();
    if (tid < 32) {
      float s = negPart[0][tid] + negPart[1][tid] +
                negPart[2][tid] + negPart[3][tid];
      ws[WS_NEG + rBase + tid] = s;
    }
  } else {
    #pragma unroll
    for (int m = 16; m >= 1; m >>= 1) {
      lossAcc += __shfl_xor(lossAcc, m, 32);
      cntAcc  += __shfl_xor(cntAcc, m, 32);
    }
    if (lane == 0) { wavePart[wave][0] = lossAcc; wavePart[wave][1] = cntAcc; }
    __syncthreads();
    if (tid == 0) {
      float L = 0.f, C = 0.f;
      for (int w = 0; w < 8; ++w) { L += wavePart[w][0]; C += wavePart[w][1]; }
      ws[WS_PART + 2 * blockIdx.x]     = L;
      ws[WS_PART + 2 * blockIdx.x + 1] = C;
    }
  }
}

// ---------------- Kernel 4: deterministic final reduction -------------------
__global__ __launch_bounds__(256)
void finalize_kernel(const float* __restrict__ ws, float* __restrict__ out) {
  __shared__ float L[256], C[256];
  int t = threadIdx.x;
  L[t] = ws[WS_PART + 2 * t];
  C[t] = ws[WS_PART + 2 * t + 1];
  __syncthreads();
  for (int s = 128; s >= 1; s >>= 1) {
    if (t < s) { L[t] += L[t + s]; C[t] += C[t + s]; }
    __syncthreads();
  }
  if (t == 0) out[0] = (C[0] > 0.f) ? (L[0] / C[0]) : 0.f;
}

extern "C" void kernel_launch(void* const* d_in, const int* in_sizes, int n_in,
                              void* d_out, int out_size, void* d_ws, size_t ws_size,
                              hipStream_t stream) {
  (void)in_sizes; (void)n_in; (void)out_size; (void)ws_size;
  const float* x   = (const float*)d_in[0];
  const int*   lab = (const int*)d_in[1];
  float*       ws  = (float*)d_ws;
  float*       out = (float*)d_out;

  norms_kernel<<<N_PTS / 8, 256, 0, stream>>>(x, ws);        // wave per row
  pass_kernel<0><<<NBLK, 256, 0, stream>>>(x, lab, ws);      // neg row sums
  pass_kernel<1><<<NBLK, 256, 0, stream>>>(x, lab, ws);      // positive terms
  finalize_kernel<<<1, 256, 0, stream>>>(ws, out);           // loss / n_pos
}